// STGAFormer_5652176962360
// MI455X (gfx1250) — compile-verified
//
#include <hip/hip_runtime.h>
#include <cstddef>

// ---------------- problem constants ----------------
static constexpr int B_  = 16;
static constexpr int T_  = 12;
static constexpr int N_  = 358;
static constexpr int D_  = 152;
static constexpr int DH_ = 76;
static constexpr int M_  = T_ * B_ * N_;      // 68736 = 32 * 2148
static constexpr int MT32 = M_ / 32;          // 2148 row tiles of 32

typedef float v2f __attribute__((ext_vector_type(2)));
typedef float v4f __attribute__((ext_vector_type(4)));
typedef float v8f __attribute__((ext_vector_type(8)));

// map folded row m -> element offset of row start inside x[B,T,N,D] (also d_out)
__device__ __forceinline__ size_t xmap(int m) {
  int tb = m / N_;
  int n  = m - tb * N_;
  int t  = tb >> 4;          // tb / B_
  int b  = tb & 15;          // tb % B_
  return ((size_t)((b * T_ + t) * N_ + n)) * D_;
}

// ---------------- prep: fold weights ----------------
__global__ void prep_kernel(const float* __restrict__ w_imp0,
                            const float* __restrict__ w_sim0,
                            const float* __restrict__ w_gat,
                            const float* __restrict__ w_mv,
                            const float* __restrict__ b_mv,
                            float* __restrict__ w_cat,
                            float* __restrict__ wgat_sum,
                            float* __restrict__ wmv_col) {
  int e = blockIdx.x * blockDim.x + threadIdx.x;
  if (e < D_ * D_) {
    int k = e / D_, j = e - k * D_;
    w_cat[e] = (j < DH_) ? w_imp0[k * DH_ + j] : w_sim0[k * DH_ + (j - DH_)];
  }
  if (e < DH_ * DH_) {
    int i = e / DH_, j = e - i * DH_;
    float s = 0.f;
    for (int r = 0; r < 4; ++r) s += w_gat[(r * DH_ + i) * DH_ + j];
    wgat_sum[e] = s;
  }
  if (e < DH_) {
    float s = b_mv[e];
    for (int si = 0; si < 32; ++si) s += w_mv[si * DH_ + e];
    wmv_col[e] = s;
  }
}

// ---------------- pack B [KDIM,NCOLS] into WMMA fragment order ----------------
// P[((ct*K4 + kstep)*32 + lane)*2 + j] = W[(kstep*4 + kh + j)*NCOLS + col], 0-padded
__global__ void pack_b(const float* __restrict__ W, float* __restrict__ P,
                       int KDIM, int NCOLS, int NT) {
  int e = blockIdx.x * blockDim.x + threadIdx.x;
  int K4 = KDIM >> 2;
  int total = NT * K4 * 64;
  if (e >= total) return;
  int j = e & 1;
  int lane = (e >> 1) & 31;
  int rest = e >> 6;                 // ct*K4 + kstep
  int kstep = rest % K4;
  int ct = rest / K4;
  int col = ct * 16 + (lane & 15);
  int k = kstep * 4 + ((lane < 16) ? 0 : 2) + j;
  P[e] = (col < NCOLS) ? W[(size_t)k * NCOLS + col] : 0.f;
}

// ---------------- dynamic threshold per folded batch ----------------
__global__ void thr_kernel(const float* __restrict__ x,
                           const float* __restrict__ w1, const float* __restrict__ b1,
                           const float* __restrict__ w2, const float* __restrict__ b2,
                           float* __restrict__ thr) {
  __shared__ float xm[D_];
  __shared__ float hh[64];
  int tb = blockIdx.x;
  int t = tb >> 4, b = tb & 15;
  const float* xb = x + ((size_t)(b * T_ + t)) * N_ * D_;
  for (int d = threadIdx.x; d < D_; d += blockDim.x) {
    float s = 0.f;
    for (int n = 0; n < N_; ++n) s += xb[(size_t)n * D_ + d];
    xm[d] = s / (float)N_;
  }
  __syncthreads();
  for (int j = threadIdx.x; j < 64; j += blockDim.x) {
    float s = b1[j];
    for (int d = 0; d < D_; ++d) s += xm[d] * w1[d * 64 + j];
    hh[j] = fmaxf(s, 0.f);
  }
  __syncthreads();
  if (threadIdx.x == 0) {
    float s = b2[0];
    for (int j = 0; j < 64; ++j) s += hh[j] * w2[j];
    thr[tb] = 1.f / (1.f + expf(-s));
  }
}

// ---------------- sim_mask row mean ----------------
__global__ void smean_kernel(const float* __restrict__ dist,
                             const float* __restrict__ thr,
                             float* __restrict__ smean) {
  int i = blockIdx.x;          // 358 blocks
  int tb = threadIdx.x;        // 192 threads; same dist row -> broadcast loads
  float th = thr[tb];
  const float* row = dist + (size_t)i * N_;
  int cnt = 0;
  for (int j = 0; j < N_; ++j) cnt += (row[j] >= th) ? 1 : 0;
  smean[(size_t)tb * N_ + i] = (float)cnt / (float)N_;
}

__global__ void cond_kernel(const float* __restrict__ smean, int* __restrict__ cond) {
  int t = threadIdx.x;
  if (t < T_) {
    const float* p = smean + (size_t)(t * B_) * N_;
    float s = 0.f;
    for (int i = 0; i < N_; ++i) s += p[i];
    cond[t] = (s > 0.f) ? 1 : 0;
  }
}

// ---------------- generic WMMA f32 16x16x4 GEMM, 32 rows per block ----------------
// MODE 0: A = x rows (gathered). cols<76: relu(+b_imp0)+sg*wmv_col -> out0;
//         cols>=76: (+b_sim0[c-76])*smean[m] -> out1
// MODE 1: A = A0[M,76];               +bias0 -> out0
// MODE 2: A = relu(cond? A0 : A1);    +bias0 -> out0 (in-place with A0 is safe)
// MODE 3: A row = [A0row | A1row];    +bias0 -> out0
// MODE 4: A = A0[M,76], N=152;        +bias0 -> d_out (remapped [B,T,N,D])
template <int KDIM, int NCOLS, int MODE>
__global__ void gemm_wmma(const float* __restrict__ A0, const float* __restrict__ A1,
                          const float* __restrict__ Bp,
                          const float* __restrict__ bias0, const float* __restrict__ bias1,
                          float* __restrict__ out0, float* __restrict__ out1,
                          const float* __restrict__ smean, const int* __restrict__ cond,
                          const float* __restrict__ gate, const float* __restrict__ wmv_col) {
  constexpr int LDK = KDIM + 4;     // +4 pad: rows 16B-aligned AND 16 distinct banks
  constexpr int K4  = KDIM / 4;
  __shared__ float As[32 * LDK];
  const int row0 = blockIdx.x * 32;
  const int tid = threadIdx.x;

  // stage 32 x KDIM A-tile into LDS in float4s (all sources 16B-aligned; the
  // MODE-3 concat boundary (76 floats) is 16B-aligned so no vec straddles it)
  for (int e4 = tid; e4 < 32 * K4; e4 += blockDim.x) {
    int r = e4 / K4;
    int d = (e4 - r * K4) * 4;
    int m = row0 + r;
    v4f a;
    if constexpr (MODE == 0) {
      a = *(const v4f*)(A0 + xmap(m) + d);
    } else if constexpr (MODE == 1 || MODE == 4) {
      a = *(const v4f*)(A0 + (size_t)m * KDIM + d);
    } else if constexpr (MODE == 2) {
      int t = (m / N_) >> 4;
      const float* src = cond[t] ? A0 : A1;
      a = *(const v4f*)(src + (size_t)m * DH_ + d);
      a[0] = fmaxf(a[0], 0.f);
      a[1] = fmaxf(a[1], 0.f);
      a[2] = fmaxf(a[2], 0.f);
      a[3] = fmaxf(a[3], 0.f);
    } else { // MODE 3: concat [imp | s3]
      const float* src = (d < DH_) ? (A0 + (size_t)m * DH_ + d)
                                   : (A1 + (size_t)m * DH_ + (d - DH_));
      a = *(const v4f*)src;
    }
    *(v4f*)(As + r * LDK + d) = a;    // ds_store_b128
  }
  __syncthreads();

  const int wv = tid >> 5;
  const int lane = tid & 31;
  const int col = wv * 16 + (lane & 15);
  const bool colok = (col < NCOLS);
  const int kh = (lane < 16) ? 0 : 2;
  const int arow = lane & 15;

  const float* bp  = Bp + (size_t)wv * K4 * 64 + (size_t)lane * 2;  // packed, branchless
  const float* a0p = As + arow * LDK + kh;
  const float* a1p = As + (16 + arow) * LDK + kh;

  v8f acc0 = {0.f, 0.f, 0.f, 0.f, 0.f, 0.f, 0.f, 0.f};
  v8f acc1 = {0.f, 0.f, 0.f, 0.f, 0.f, 0.f, 0.f, 0.f};
#pragma unroll 2
  for (int k4 = 0; k4 < K4; ++k4) {
    v2f bv = *(const v2f*)(bp + (size_t)k4 * 64);   // one global_load_b64 per step
    v2f av0, av1;
    av0[0] = a0p[k4 * 4];
    av0[1] = a0p[k4 * 4 + 1];
    av1[0] = a1p[k4 * 4];
    av1[1] = a1p[k4 * 4 + 1];
    acc0 = __builtin_amdgcn_wmma_f32_16x16x4_f32(false, av0, false, bv,
                                                 (short)0, acc0, false, false);
    acc1 = __builtin_amdgcn_wmma_f32_16x16x4_f32(false, av1, false, bv,
                                                 (short)0, acc1, false, false);
  }

  if (!colok) return;
  const int rb = (lane < 16) ? 0 : 8;
  v8f accs[2] = {acc0, acc1};
#pragma unroll
  for (int h = 0; h < 2; ++h) {
#pragma unroll
    for (int v = 0; v < 8; ++v) {
      int m = row0 + h * 16 + rb + v;
      float val = accs[h][v];
      if constexpr (MODE == 0) {
        float sg = 1.f / (1.f + expf(-gate[0]));
        if (col < DH_) {
          out0[(size_t)m * DH_ + col] = fmaxf(val + bias0[col], 0.f) + sg * wmv_col[col];
        } else {
          int c = col - DH_;
          out1[(size_t)m * DH_ + c] = (val + bias1[c]) * smean[m];
        }
      } else if constexpr (MODE == 4) {
        out0[xmap(m) + col] = val + bias0[col];
      } else {
        out0[(size_t)m * DH_ + col] = val + bias0[col];
      }
    }
  }
}

// ---------------- fusion: LayerNorm + sigmoid gating (one wave per row) ----------------
__global__ void fusion_kernel(float* __restrict__ fg,       // in: fg pre-LN, out: z
                              const float* __restrict__ imp,
                              const float* __restrict__ s3,
                              const float* __restrict__ ln_g,
                              const float* __restrict__ ln_b,
                              const float* __restrict__ fw) {
  int lane = threadIdx.x & 31;
  int wv = threadIdx.x >> 5;
  int m = blockIdx.x * 8 + wv;
  if (m >= M_) return;
  float a  = 1.f / (1.f + expf(-fw[0]));
  float b2 = 1.f / (1.f + expf(-fw[1]));
  float alpha = a / (a + b2);
  const size_t base = (size_t)m * DH_;
  float v[3];
  int c[3];
  float s = 0.f, sq = 0.f;
  for (int q = 0; q < 3; ++q) {
    c[q] = lane + 32 * q;
    v[q] = (c[q] < DH_) ? fg[base + c[q]] : 0.f;
    s += v[q];
    sq += v[q] * v[q];
  }
  for (int off = 16; off; off >>= 1) {
    s  += __shfl_xor(s, off, 32);
    sq += __shfl_xor(sq, off, 32);
  }
  float mu  = s / (float)DH_;
  float var = sq / (float)DH_ - mu * mu;
  float rs  = rsqrtf(var + 1e-5f);
  for (int q = 0; q < 3; ++q) {
    if (c[q] < DH_) {
      float fn = (v[q] - mu) * rs * ln_g[c[q]] + ln_b[c[q]];
      float g = 1.f / (1.f + expf(-fn));
      float comb = alpha * imp[base + c[q]] + (1.f - alpha) * s3[base + c[q]];
      fg[base + c[q]] = g * comb + (1.f - g) * g;
    }
  }
}

// ---------------- host side ----------------
static inline size_t psize(int KDIM, int NCOLS) {   // packed B size in floats
  int NT = (NCOLS + 15) / 16;
  return (size_t)NT * KDIM * 16;
}

extern "C" void kernel_launch(void* const* d_in, const int* in_sizes, int n_in,
                              void* d_out, int out_size, void* d_ws, size_t ws_size,
                              hipStream_t stream) {
  (void)in_sizes; (void)n_in; (void)out_size; (void)ws_size;
  const float* x      = (const float*)d_in[0];
  const float* dist   = (const float*)d_in[1];
  const float* w_thr1 = (const float*)d_in[2];
  const float* b_thr1 = (const float*)d_in[3];
  const float* w_thr2 = (const float*)d_in[4];
  const float* b_thr2 = (const float*)d_in[5];
  const float* w_imp0 = (const float*)d_in[6];
  const float* b_imp0 = (const float*)d_in[7];
  // d_in[8] w_mk, d_in[9] b_mk: algebraically dead (softmax rows sum to 1)
  const float* w_mv   = (const float*)d_in[10];
  const float* b_mv   = (const float*)d_in[11];
  const float* gate   = (const float*)d_in[12];
  const float* w_sim0 = (const float*)d_in[13];
  const float* b_sim0 = (const float*)d_in[14];
  const float* w_gat  = (const float*)d_in[15];
  const float* b_gat  = (const float*)d_in[16];
  const float* w_sim3 = (const float*)d_in[17];
  const float* b_sim3 = (const float*)d_in[18];
  const float* w_fuse = (const float*)d_in[19];
  const float* b_fuse = (const float*)d_in[20];
  const float* ln_g   = (const float*)d_in[21];
  const float* ln_b   = (const float*)d_in[22];
  const float* w_proj = (const float*)d_in[23];
  const float* b_proj = (const float*)d_in[24];
  const float* fw     = (const float*)d_in[25];

  char* ws = (char*)d_ws;
  size_t off = 0;
  auto take = [&](size_t nbytes) -> char* {
    char* p = ws + off;
    off = (off + nbytes + 255) & ~(size_t)255;
    return p;
  };
  float* impbuf  = (float*)take((size_t)M_ * DH_ * 4); // imp_feat
  float* sbuf    = (float*)take((size_t)M_ * DH_ * 4); // s1 -> (reused) fg/z
  float* gbuf    = (float*)take((size_t)M_ * DH_ * 4); // gat -> s3 (in place)
  float* smean   = (float*)take((size_t)M_ * 4);
  float* thr     = (float*)take(192 * 4);
  int*   cond    = (int*)  take(T_ * 4);
  float* w_cat   = (float*)take((size_t)D_ * D_ * 4);
  float* wgatsum = (float*)take((size_t)DH_ * DH_ * 4);
  float* wmv_col = (float*)take(DH_ * 4);
  float* p_cat   = (float*)take(psize(152, 152) * 4);  // packed B matrices
  float* p_gat   = (float*)take(psize(76, 76) * 4);
  float* p_sim3  = (float*)take(psize(76, 76) * 4);
  float* p_fuse  = (float*)take(psize(152, 76) * 4);
  float* p_proj  = (float*)take(psize(76, 152) * 4);

  // 0) fold weights, then pack all B matrices into WMMA fragment order
  prep_kernel<<<(D_ * D_ + 255) / 256, 256, 0, stream>>>(
      w_imp0, w_sim0, w_gat, w_mv, b_mv, w_cat, wgatsum, wmv_col);
  auto launch_pack = [&](const float* W, float* P, int K, int NC) {
    int NT = (NC + 15) / 16;
    int total = NT * (K / 4) * 64;
    pack_b<<<(total + 255) / 256, 256, 0, stream>>>(W, P, K, NC, NT);
  };
  launch_pack(w_cat,   p_cat,  152, 152);
  launch_pack(wgatsum, p_gat,   76,  76);
  launch_pack(w_sim3,  p_sim3,  76,  76);
  launch_pack(w_fuse,  p_fuse, 152,  76);
  launch_pack(w_proj,  p_proj,  76, 152);

  // 1) thresholds, 2) sim-mask row means + cond
  thr_kernel<<<T_ * B_, 256, 0, stream>>>(x, w_thr1, b_thr1, w_thr2, b_thr2, thr);
  smean_kernel<<<N_, T_ * B_, 0, stream>>>(dist, thr, smean);
  cond_kernel<<<1, 32, 0, stream>>>(smean, cond);

  // 3) G1: x -> imp_feat & s1                                 [K=152, N=152]
  gemm_wmma<152, 152, 0><<<MT32, 320, 0, stream>>>(
      x, nullptr, p_cat, b_imp0, b_sim0, impbuf, sbuf, smean, nullptr, gate, wmv_col);
  // 4) G2: gat = s1 @ wgat_sum + b_gat                        [76 -> 76]
  gemm_wmma<76, 76, 1><<<MT32, 160, 0, stream>>>(
      sbuf, nullptr, p_gat, b_gat, nullptr, gbuf, nullptr, nullptr, nullptr, nullptr, nullptr);
  // 5) G3: s3 = relu(cond? gat : s1) @ w_sim3 + b_sim3 (in-place gbuf)
  gemm_wmma<76, 76, 2><<<MT32, 160, 0, stream>>>(
      gbuf, sbuf, p_sim3, b_sim3, nullptr, gbuf, nullptr, nullptr, cond, nullptr, nullptr);
  // 6) G4: fg = [imp|s3] @ w_fuse + b_fuse -> sbuf            [K=152 -> 76]
  gemm_wmma<152, 76, 3><<<MT32, 160, 0, stream>>>(
      impbuf, gbuf, p_fuse, b_fuse, nullptr, sbuf, nullptr, nullptr, nullptr, nullptr, nullptr);
  // 7) fusion: LN + gating, z in-place in sbuf
  fusion_kernel<<<M_ / 8, 256, 0, stream>>>(sbuf, impbuf, gbuf, ln_g, ln_b, fw);
  // 8) G5: out = z @ w_proj + b_proj -> [B,T,N,D]             [76 -> 152]
  gemm_wmma<76, 152, 4><<<MT32, 320, 0, stream>>>(
      sbuf, nullptr, p_proj, b_proj, nullptr, (float*)d_out, nullptr, nullptr, nullptr, nullptr, nullptr);
}